// CLS_30562987278491
// MI455X (gfx1250) — compile-verified
//
#include <hip/hip_runtime.h>
#include <math.h>

typedef float v2f __attribute__((ext_vector_type(2)));
typedef float v8f __attribute__((ext_vector_type(8)));

#define V8Z {0.f,0.f,0.f,0.f,0.f,0.f,0.f,0.f}
#define TWO_PI 6.283185307179586f

__device__ __forceinline__ v8f wmma_f32(v2f a, v2f b, v8f c) {
  // D = A(16x4) * B(4x16) + C(16x16), fp32 WMMA
  return __builtin_amdgcn_wmma_f32_16x16x4_f32(false, a, false, b, (short)0, c, false, false);
}

// ---------------------------------------------------------------------------
// constants
#define NS   288
#define NS2  82944          // 288*288
#define PIX  60516          // 246*246
#define NT16 3783           // ceil(60516/16)

// ---------------------------------------------------------------------------
// DFT matrix: Fre/Fim[j*288+k] = cos/sin(-2*pi*j*k/288)
__global__ void k_dftmat(float* __restrict__ Fre, float* __restrict__ Fim) {
  int t = blockIdx.x * blockDim.x + threadIdx.x;
  if (t >= NS2) return;
  int j = t / NS, k = t % NS;
  int m = (int)(((long)j * k) % NS);
  float ang = -TWO_PI * (float)m / (float)NS;
  float s, c; sincosf(ang, &s, &c);
  Fre[t] = c; Fim[t] = s;
}

// ---------------------------------------------------------------------------
// 1x1 conv 64->16 as WMMA GEMM: cls[b,co,p] = sum_ci w[co,ci]*x[b,ci,p] + bias[co]
__global__ void k_conv_red(const float* __restrict__ x, const float* __restrict__ w,
                           const float* __restrict__ bias, float* __restrict__ out) {
  int wave = (blockIdx.x * blockDim.x + threadIdx.x) >> 5;
  if (wave >= 8 * NT16) return;
  int b = wave / NT16, nt = wave % NT16;
  int p0 = nt * 16;
  int lane = threadIdx.x & 31, half = lane >> 4, l16 = lane & 15;
  int p = p0 + l16;
  int pc = p < PIX ? p : PIX - 1;
  const float* xb = x + (long)b * 64 * PIX;
  v8f acc = V8Z;
  for (int k0 = 0; k0 < 64; k0 += 4) {
    int ka = k0 + 2 * half;
    v2f a; a.x = w[l16 * 64 + ka];          a.y = w[l16 * 64 + ka + 1];
    v2f bb; bb.x = xb[(long)ka * PIX + pc]; bb.y = xb[(long)(ka + 1) * PIX + pc];
    acc = wmma_f32(a, bb, acc);
  }
  if (p < PIX) {
    float* ob = out + (long)b * 16 * PIX + p;
    for (int v = 0; v < 8; v++) {
      int co = v + 8 * half;
      ob[(long)co * PIX] = acc[v] + bias[co];
    }
  }
}

// ---------------------------------------------------------------------------
// 3x3 conv 16->16, VALID, optional leaky. One thread per (b,y,x), all 16 co.
__global__ void k_conv3x3(const float* __restrict__ in, const float* __restrict__ w,
                          const float* __restrict__ bias, float* __restrict__ out,
                          int Hin, int leaky_flag) {
  int Hout = Hin - 2;
  long total = 8L * Hout * Hout;
  long t = (long)blockIdx.x * blockDim.x + threadIdx.x;
  if (t >= total) return;
  int xx = (int)(t % Hout);
  int yy = (int)((t / Hout) % Hout);
  int b  = (int)(t / ((long)Hout * Hout));
  float acc[16];
  for (int co = 0; co < 16; co++) acc[co] = bias[co];
  const float* inb = in + (long)b * 16 * Hin * Hin;
  for (int ci = 0; ci < 16; ci++) {
    const float* p = inb + (long)ci * Hin * Hin + (long)yy * Hin + xx;
    float v0 = p[0],        v1 = p[1],          v2 = p[2];
    float v3 = p[Hin],      v4 = p[Hin + 1],    v5 = p[Hin + 2];
    float v6 = p[2 * Hin],  v7 = p[2 * Hin + 1], v8v = p[2 * Hin + 2];
    for (int co = 0; co < 16; co++) {
      const float* wp = w + (co * 16 + ci) * 9;
      acc[co] += v0*wp[0] + v1*wp[1] + v2*wp[2] + v3*wp[3] + v4*wp[4]
               + v5*wp[5] + v6*wp[6] + v7*wp[7] + v8v*wp[8];
    }
  }
  float* ob = out + (long)b * 16 * Hout * Hout + (long)yy * Hout + xx;
  for (int co = 0; co < 16; co++) {
    float v = acc[co];
    if (leaky_flag) v = (v >= 0.f) ? v : 0.1f * v;
    ob[(long)co * Hout * Hout] = v;
  }
}

// ---------------------------------------------------------------------------
// adaptive 3x3 avg pool on (8,16,240,240): one block per (b,c,i,j)
__global__ void k_pool(const float* __restrict__ h, float* __restrict__ pooled) {
  int blk = blockIdx.x;                       // ((b*16+c)*3+i)*3+j
  int j = blk % 3, i = (blk / 3) % 3, c = (blk / 9) % 16, b = blk / 144;
  const float* base = h + (long)(b * 16 + c) * 57600 + (long)i * 80 * 240 + j * 80;
  float s = 0.f;
  for (int t = threadIdx.x; t < 6400; t += 256) {
    int yy = t / 80, xx = t % 80;
    s += base[(long)yy * 240 + xx];
  }
  __shared__ float sm[256];
  sm[threadIdx.x] = s; __syncthreads();
  for (int off = 128; off > 0; off >>= 1) {
    if ((int)threadIdx.x < off) sm[threadIdx.x] += sm[threadIdx.x + off];
    __syncthreads();
  }
  if (threadIdx.x == 0) pooled[blk] = sm[0] / 6400.f;
}

// ---------------------------------------------------------------------------
// kernel_P = exp(1x1 conv) - spatial mean. One thread per (b,co).
__global__ void k_kernelP(const float* __restrict__ pooled, const float* __restrict__ w4,
                          const float* __restrict__ b4, float* __restrict__ kp) {
  int t = blockIdx.x * blockDim.x + threadIdx.x;
  if (t >= 128) return;
  int b = t / 16, co = t % 16;
  float e[9]; float mean = 0.f;
  for (int s = 0; s < 9; s++) {
    float v = b4[co];
    for (int ci = 0; ci < 16; ci++) v += w4[co * 16 + ci] * pooled[(b * 16 + ci) * 9 + s];
    e[s] = expf(v); mean += e[s];
  }
  mean /= 9.f;
  for (int s = 0; s < 9; s++) kp[(b * 16 + co) * 9 + s] = e[s] - mean;
}

// ---------------------------------------------------------------------------
// edge pad cls (246x246) -> fp (288x288) for one batch
__global__ void k_pad(const float* __restrict__ cls, float* __restrict__ fp, int b) {
  long t = (long)blockIdx.x * blockDim.x + threadIdx.x;
  if (t >= 16L * NS2) return;
  int x = (int)(t % NS);
  int y = (int)((t / NS) % NS);
  int c = (int)(t / NS2);
  int iy = y - 21; iy = iy < 0 ? 0 : (iy > 245 ? 245 : iy);
  int ix = x - 21; ix = ix < 0 ? 0 : (ix > 245 ? 245 : ix);
  fp[t] = cls[(long)(b * 16 + c) * PIX + (long)iy * 246 + ix];
}

// ---------------------------------------------------------------------------
// analytic OTF of 21x21 blur kernel, two-stage twiddle sum (per batch)
__global__ void k_kf1(const float* __restrict__ ker, float* __restrict__ R, int b) {
  int t = blockIdx.x * blockDim.x + threadIdx.x;
  if (t >= 21 * NS) return;
  int a = t / NS, v = t % NS;
  float re = 0.f, im = 0.f;
  for (int j = 0; j < 21; j++) {
    int m = (v * (j - 10)) % NS; if (m < 0) m += NS;
    float ang = -TWO_PI * (float)m / (float)NS;
    float s, c; sincosf(ang, &s, &c);
    float kv = ker[b * 441 + a * 21 + j];
    re += kv * c; im += kv * s;
  }
  R[2 * t] = re; R[2 * t + 1] = im;
}

__global__ void k_kf2(const float* __restrict__ R, float* __restrict__ Kre,
                      float* __restrict__ Kim) {
  int t = blockIdx.x * blockDim.x + threadIdx.x;
  if (t >= NS2) return;
  int u = t / NS, v = t % NS;
  float re = 0.f, im = 0.f;
  for (int a = 0; a < 21; a++) {
    int m = (u * (a - 10)) % NS; if (m < 0) m += NS;
    float ang = -TWO_PI * (float)m / (float)NS;
    float s, c; sincosf(ang, &s, &c);
    float rr = R[2 * (a * NS + v)], ri = R[2 * (a * NS + v) + 1];
    re += c * rr - s * ri;
    im += c * ri + s * rr;
  }
  Kre[t] = re; Kim[t] = im;
}

// ---------------------------------------------------------------------------
// FFT forward stage 1: T = F * X  (X real), per batch, 16 images
__global__ void k_fft_fwd1(const float* __restrict__ Fre, const float* __restrict__ Fim,
                           const float* __restrict__ X, float* __restrict__ Tre,
                           float* __restrict__ Tim) {
  int wave = (blockIdx.x * blockDim.x + threadIdx.x) >> 5;
  if (wave >= 16 * 324) return;
  int img = wave / 324, t = wave % 324;
  int m0 = (t / 18) * 16, n0 = (t % 18) * 16;
  int lane = threadIdx.x & 31, half = lane >> 4, l16 = lane & 15;
  const float* Xi = X + (long)img * NS2;
  v8f ar = V8Z, ai = V8Z;
  for (int k0 = 0; k0 < NS; k0 += 4) {
    int ka = k0 + 2 * half;
    v2f bb; bb.x = Xi[(long)ka * NS + n0 + l16]; bb.y = Xi[(long)(ka + 1) * NS + n0 + l16];
    v2f fa; fa.x = Fre[(m0 + l16) * NS + ka];    fa.y = Fre[(m0 + l16) * NS + ka + 1];
    v2f fb; fb.x = Fim[(m0 + l16) * NS + ka];    fb.y = Fim[(m0 + l16) * NS + ka + 1];
    ar = wmma_f32(fa, bb, ar);
    ai = wmma_f32(fb, bb, ai);
  }
  float* tr = Tre + (long)img * NS2;
  float* ti = Tim + (long)img * NS2;
  for (int v = 0; v < 8; v++) {
    int row = m0 + v + 8 * half, col = n0 + l16;
    tr[row * NS + col] = ar[v];
    ti[row * NS + col] = ai[v];
  }
}

// FFT forward stage 2: Y = T * F  (complex x complex, F symmetric)
__global__ void k_fft_fwd2(const float* __restrict__ Fre, const float* __restrict__ Fim,
                           const float* __restrict__ Tre, const float* __restrict__ Tim,
                           float* __restrict__ Gre, float* __restrict__ Gim) {
  int wave = (blockIdx.x * blockDim.x + threadIdx.x) >> 5;
  if (wave >= 16 * 324) return;
  int img = wave / 324, t = wave % 324;
  int m0 = (t / 18) * 16, n0 = (t % 18) * 16;
  int lane = threadIdx.x & 31, half = lane >> 4, l16 = lane & 15;
  const float* Ar = Tre + (long)img * NS2;
  const float* Ai = Tim + (long)img * NS2;
  v8f yr = V8Z, yi = V8Z;
  for (int k0 = 0; k0 < NS; k0 += 4) {
    int ka = k0 + 2 * half;
    v2f ar; ar.x = Ar[(m0 + l16) * NS + ka]; ar.y = Ar[(m0 + l16) * NS + ka + 1];
    v2f ai; ai.x = Ai[(m0 + l16) * NS + ka]; ai.y = Ai[(m0 + l16) * NS + ka + 1];
    v2f br; br.x = Fre[ka * NS + n0 + l16];  br.y = Fre[(ka + 1) * NS + n0 + l16];
    v2f bi; bi.x = Fim[ka * NS + n0 + l16];  bi.y = Fim[(ka + 1) * NS + n0 + l16];
    yr = wmma_f32(ar, br, yr);
    v2f ain; ain.x = -ai.x; ain.y = -ai.y;
    yr = wmma_f32(ain, bi, yr);              // - Tim*Fim
    yi = wmma_f32(ar, bi, yi);
    yi = wmma_f32(ai, br, yi);
  }
  float* gr = Gre + (long)img * NS2;
  float* gi = Gim + (long)img * NS2;
  for (int v = 0; v < 8; v++) {
    int row = m0 + v + 8 * half, col = n0 + l16;
    gr[row * NS + col] = yr[v];
    gi[row * NS + col] = yi[v];
  }
}

// apply inverse Wiener filter in frequency domain (per batch, in place on G)
__global__ void k_apply(const float* __restrict__ kp, const float* __restrict__ Kre,
                        const float* __restrict__ Kim, float* __restrict__ Gre,
                        float* __restrict__ Gim, int b) {
  long t = (long)blockIdx.x * blockDim.x + threadIdx.x;
  if (t >= 16L * NS2) return;
  int c = (int)(t / NS2);
  int u = (int)((t / NS) % NS);
  int v = (int)(t % NS);
  const float* k9 = kp + (b * 16 + c) * 9;
  float pre = 0.f, pim = 0.f;
  for (int a = 0; a < 3; a++)
    for (int d = 0; d < 3; d++) {
      int m = (u * (a - 1) + v * (d - 1)) % NS; if (m < 0) m += NS;
      float ang = -TWO_PI * (float)m / (float)NS;
      float s, cc; sincosf(ang, &s, &cc);
      float kv = k9[a * 3 + d];
      pre += kv * cc; pim += kv * s;
    }
  float kr = Kre[u * NS + v], ki = Kim[u * NS + v];
  float denom = kr * kr + ki * ki + pre * pre + pim * pim;
  float gr = Gre[t], gi = Gim[t];
  Gre[t] = (kr * gr + ki * gi) / denom;   // conj(K)*G / denom
  Gim[t] = (kr * gi - ki * gr) / denom;
}

// FFT inverse stage 1: U = conj(F) * G
__global__ void k_fft_inv1(const float* __restrict__ Fre, const float* __restrict__ Fim,
                           const float* __restrict__ Gre, const float* __restrict__ Gim,
                           float* __restrict__ Ure, float* __restrict__ Uim) {
  int wave = (blockIdx.x * blockDim.x + threadIdx.x) >> 5;
  if (wave >= 16 * 324) return;
  int img = wave / 324, t = wave % 324;
  int m0 = (t / 18) * 16, n0 = (t % 18) * 16;
  int lane = threadIdx.x & 31, half = lane >> 4, l16 = lane & 15;
  const float* Br = Gre + (long)img * NS2;
  const float* Bi = Gim + (long)img * NS2;
  v8f ur = V8Z, ui = V8Z;
  for (int k0 = 0; k0 < NS; k0 += 4) {
    int ka = k0 + 2 * half;
    v2f afr; afr.x = Fre[(m0 + l16) * NS + ka]; afr.y = Fre[(m0 + l16) * NS + ka + 1];
    v2f afi; afi.x = Fim[(m0 + l16) * NS + ka]; afi.y = Fim[(m0 + l16) * NS + ka + 1];
    v2f bgr; bgr.x = Br[ka * NS + n0 + l16];    bgr.y = Br[(ka + 1) * NS + n0 + l16];
    v2f bgi; bgi.x = Bi[ka * NS + n0 + l16];    bgi.y = Bi[(ka + 1) * NS + n0 + l16];
    ur = wmma_f32(afr, bgr, ur);
    ur = wmma_f32(afi, bgi, ur);               // + Fim*Gim
    ui = wmma_f32(afr, bgi, ui);
    v2f afin; afin.x = -afi.x; afin.y = -afi.y;
    ui = wmma_f32(afin, bgr, ui);              // - Fim*Gre
  }
  float* tr = Ure + (long)img * NS2;
  float* ti = Uim + (long)img * NS2;
  for (int v = 0; v < 8; v++) {
    int row = m0 + v + 8 * half, col = n0 + l16;
    tr[row * NS + col] = ur[v];
    ti[row * NS + col] = ui[v];
  }
}

// FFT inverse stage 2 (real output): clear = Re(U * conj(F)) / N^2
__global__ void k_fft_inv2(const float* __restrict__ Fre, const float* __restrict__ Fim,
                           const float* __restrict__ Ure, const float* __restrict__ Uim,
                           float* __restrict__ outp) {
  int wave = (blockIdx.x * blockDim.x + threadIdx.x) >> 5;
  if (wave >= 16 * 324) return;
  int img = wave / 324, t = wave % 324;
  int m0 = (t / 18) * 16, n0 = (t % 18) * 16;
  int lane = threadIdx.x & 31, half = lane >> 4, l16 = lane & 15;
  const float* Ar = Ure + (long)img * NS2;
  const float* Ai = Uim + (long)img * NS2;
  v8f acc = V8Z;
  for (int k0 = 0; k0 < NS; k0 += 4) {
    int ka = k0 + 2 * half;
    v2f aur; aur.x = Ar[(m0 + l16) * NS + ka]; aur.y = Ar[(m0 + l16) * NS + ka + 1];
    v2f aui; aui.x = Ai[(m0 + l16) * NS + ka]; aui.y = Ai[(m0 + l16) * NS + ka + 1];
    v2f bfr; bfr.x = Fre[ka * NS + n0 + l16];  bfr.y = Fre[(ka + 1) * NS + n0 + l16];
    v2f bfi; bfi.x = Fim[ka * NS + n0 + l16];  bfi.y = Fim[(ka + 1) * NS + n0 + l16];
    acc = wmma_f32(aur, bfr, acc);             // Ure*Fre
    acc = wmma_f32(aui, bfi, acc);             // + Uim*Fim  (Re of U*conj(F))
  }
  float* op = outp + (long)img * NS2;
  const float scale = 1.0f / (float)NS2;
  for (int v = 0; v < 8; v++) {
    int row = m0 + v + 8 * half, col = n0 + l16;
    op[row * NS + col] = acc[v] * scale;
  }
}

// ---------------------------------------------------------------------------
// final 1x1 conv 16->64 as WMMA GEMM, reading the cropped deblur (per batch)
__global__ void k_conv_exp(const float* __restrict__ cf, const float* __restrict__ w,
                           const float* __restrict__ bias, float* __restrict__ out) {
  int wave = (blockIdx.x * blockDim.x + threadIdx.x) >> 5;
  if (wave >= 4 * NT16) return;
  int mt = wave / NT16, nt = wave % NT16;
  int m0 = mt * 16, p0 = nt * 16;
  int lane = threadIdx.x & 31, half = lane >> 4, l16 = lane & 15;
  int p = p0 + l16;
  int pc = p < PIX ? p : PIX - 1;
  int yy = pc / 246, xx = pc % 246;
  const float* bsrc = cf + (long)(yy + 21) * NS + (xx + 21);   // crop offset
  v8f acc = V8Z;
  for (int k0 = 0; k0 < 16; k0 += 4) {
    int ka = k0 + 2 * half;
    v2f a; a.x = w[(m0 + l16) * 16 + ka];  a.y = w[(m0 + l16) * 16 + ka + 1];
    v2f bb; bb.x = bsrc[(long)ka * NS2];   bb.y = bsrc[(long)(ka + 1) * NS2];
    acc = wmma_f32(a, bb, acc);
  }
  if (p < PIX) {
    for (int v = 0; v < 8; v++) {
      int co = m0 + v + 8 * half;
      out[(long)co * PIX + p] = acc[v] + bias[co];
    }
  }
}

// ---------------------------------------------------------------------------
extern "C" void kernel_launch(void* const* d_in, const int* in_sizes, int n_in,
                              void* d_out, int out_size, void* d_ws, size_t ws_size,
                              hipStream_t stream) {
  const float* x      = (const float*)d_in[0];
  const float* kernel = (const float*)d_in[1];
  const float* w_red  = (const float*)d_in[2];
  const float* b_red  = (const float*)d_in[3];
  const float* w_g1   = (const float*)d_in[4];
  const float* b_g1   = (const float*)d_in[5];
  const float* w_g2   = (const float*)d_in[6];
  const float* b_g2   = (const float*)d_in[7];
  const float* w_g3   = (const float*)d_in[8];
  const float* b_g3   = (const float*)d_in[9];
  const float* w_g4   = (const float*)d_in[10];
  const float* b_g4   = (const float*)d_in[11];
  const float* w_exp  = (const float*)d_in[12];
  const float* b_exp  = (const float*)d_in[13];
  float* out = (float*)d_out;
  float* ws  = (float*)d_ws;

  // workspace layout (floats)
  const long OFF_FRE  = 0;
  const long OFF_FIM  = OFF_FRE + NS2;
  const long OFF_CLS  = OFF_FIM + NS2;              // 128*60516
  const long OFF_HA   = OFF_CLS + 128L * 60516;     // 128*59536
  const long OFF_HB   = OFF_HA  + 128L * 59536;     // 128*58564
  const long OFF_POOL = OFF_HB  + 128L * 58564;     // 1152
  const long OFF_KP   = OFF_POOL + 1152;            // 1152
  const long OFF_R    = OFF_KP   + 1152;            // 21*288*2
  const long OFF_KRE  = OFF_R    + 21L * NS * 2;    // NS2
  const long OFF_KIM  = OFF_KRE  + NS2;             // NS2
  const long OFF_FP   = OFF_KIM  + NS2;             // 16*NS2 (pad buf, later clearFull)
  const long OFF_TRE  = OFF_FP   + 16L * NS2;
  const long OFF_TIM  = OFF_TRE  + 16L * NS2;
  const long OFF_GRE  = OFF_TIM  + 16L * NS2;
  const long OFF_GIM  = OFF_GRE  + 16L * NS2;

  float* Fre  = ws + OFF_FRE;
  float* Fim  = ws + OFF_FIM;
  float* cls  = ws + OFF_CLS;
  float* hA   = ws + OFF_HA;
  float* hB   = ws + OFF_HB;
  float* pool = ws + OFF_POOL;
  float* kp   = ws + OFF_KP;
  float* Rbuf = ws + OFF_R;
  float* Kre  = ws + OFF_KRE;
  float* Kim  = ws + OFF_KIM;
  float* fp   = ws + OFF_FP;
  float* Tre  = ws + OFF_TRE;
  float* Tim  = ws + OFF_TIM;
  float* Gre  = ws + OFF_GRE;
  float* Gim  = ws + OFF_GIM;

  // DFT matrix
  k_dftmat<<<(NS2 + 255) / 256, 256, 0, stream>>>(Fre, Fim);

  // cls_feats = 1x1 conv 64->16 (WMMA), all batches
  k_conv_red<<<NT16, 256, 0, stream>>>(x, w_red, b_red, cls);

  // 3x3 conv chain
  {
    long n1 = 8L * 244 * 244;
    k_conv3x3<<<(unsigned)((n1 + 255) / 256), 256, 0, stream>>>(cls, w_g1, b_g1, hA, 246, 1);
    long n2 = 8L * 242 * 242;
    k_conv3x3<<<(unsigned)((n2 + 255) / 256), 256, 0, stream>>>(hA, w_g2, b_g2, hB, 244, 1);
    long n3 = 8L * 240 * 240;
    k_conv3x3<<<(unsigned)((n3 + 255) / 256), 256, 0, stream>>>(hB, w_g3, b_g3, hA, 242, 0);
  }

  // pool + kernel_P
  k_pool<<<1152, 256, 0, stream>>>(hA, pool);
  k_kernelP<<<1, 128, 0, stream>>>(pool, w_g4, b_g4, kp);

  // per-batch FFT Wiener deconvolution + final conv
  const int gElem = (int)((16L * NS2 + 255) / 256);   // 5184
  const int gGemm = (16 * 324) / 8;                   // 648 blocks of 8 waves
  for (int b = 0; b < 8; b++) {
    k_pad<<<gElem, 256, 0, stream>>>(cls, fp, b);
    k_kf1<<<(21 * NS + 255) / 256, 256, 0, stream>>>(kernel, Rbuf, b);
    k_kf2<<<(NS2 + 255) / 256, 256, 0, stream>>>(Rbuf, Kre, Kim);
    k_fft_fwd1<<<gGemm, 256, 0, stream>>>(Fre, Fim, fp, Tre, Tim);
    k_fft_fwd2<<<gGemm, 256, 0, stream>>>(Fre, Fim, Tre, Tim, Gre, Gim);
    k_apply<<<gElem, 256, 0, stream>>>(kp, Kre, Kim, Gre, Gim, b);
    k_fft_inv1<<<gGemm, 256, 0, stream>>>(Fre, Fim, Gre, Gim, Tre, Tim);
    k_fft_inv2<<<gGemm, 256, 0, stream>>>(Fre, Fim, Tre, Tim, fp);
    k_conv_exp<<<(4 * NT16 + 7) / 8, 256, 0, stream>>>(fp, w_exp, b_exp,
                                                       out + (long)b * 64 * PIX);
  }
}